// Relut_35562329211101
// MI455X (gfx1250) — compile-verified
//
#include <hip/hip_runtime.h>

// relut(P) for B*C = 2048 SPD matrices of size 128x128, all-GEMM formulation:
//   Gs,Gi via coupled Newton-Schulz sqrt iteration; log/exp via Horner series.
// One workgroup (8 wave32) per matrix, everything resident in LDS, FP32 WMMA.
// Tile in/out moved by the Tensor Data Mover (tensor_load_to_lds /
// tensor_store_from_lds), with hardware row-padding for conflict-free pitch.

#define LD     132          // padded LDS row pitch (floats) -> conflict-free col gathers
#define ALPHA  1e-5f        // add_id_matrix alpha from the reference
#define NS_ITERS 18
#define LOG_TERMS 8
#define EXP_TERMS 8

typedef float v2f __attribute__((ext_vector_type(2)));
typedef float v8f __attribute__((ext_vector_type(8)));
typedef unsigned int u32;
typedef u32 v4u __attribute__((ext_vector_type(4)));
typedef u32 v8u __attribute__((ext_vector_type(8)));

// ---- Tensor Data Mover: 128x128 fp32 tile, global <-> LDS -------------------
// D# group0: [1:0]=count=1, [63:32]=lds_addr, [120:64]=global_addr, [127:126]=type=2
// D# group1: [17:16]=data_size(2=4B), [20]=pad_enable, [24:22]=pad_interval,
//            [31:25]=pad_amount, tensor_dim0/1=128, tile_dim0/1=128, dim0_stride=128
__device__ __forceinline__ void tdm_tile_desc(v4u& g0, v8u& g1,
                                              const void* gaddr, void* laddr,
                                              int padded) {
  const u32 lds = (u32)(uintptr_t)laddr;                 // LDS byte offset
  const unsigned long long ga = (unsigned long long)(uintptr_t)gaddr;
  g0[0] = 1u;                                            // count=1, user descriptor
  g0[1] = lds;
  g0[2] = (u32)ga;
  g0[3] = (u32)(ga >> 32) | 0x80000000u;                 // addr[56:32] | type=2
  // pad_interval=6 -> every 128 dwords; pad_amount=3 -> 4 dwords (pitch 132)
  g1[0] = (2u << 16) | (padded ? ((1u << 20) | (6u << 22) | (3u << 25)) : 0u);
  g1[1] = (128u << 16);                                  // tensor_dim0 = 128
  g1[2] = (128u << 16);                                  // tensor_dim1 = 128
  g1[3] = (128u << 16);                                  // tile_dim0 = 128
  g1[4] = 128u;                                          // tile_dim1 = 128, tile_dim2 = 0
  g1[5] = 128u;                                          // tensor_dim0_stride = 128
  g1[6] = 0u;
  g1[7] = 0u;
}

__device__ __forceinline__ void tdm_load_tile(const float* gsrc, float* ldsDst) {
  v4u g0; v8u g1;
  tdm_tile_desc(g0, g1, gsrc, ldsDst, /*padded=*/1);
  const v4u gz = {0u, 0u, 0u, 0u};
  asm volatile("tensor_load_to_lds %0, %1, %2, %3"
               :: "s"(g0), "s"(g1), "s"(gz), "s"(gz) : "memory");
}

__device__ __forceinline__ void tdm_store_tile(float* gdst, const float* ldsSrc) {
  v4u g0; v8u g1;
  tdm_tile_desc(g0, g1, gdst, (void*)ldsSrc, /*padded=*/0);  // no de-pad on store
  const v4u gz = {0u, 0u, 0u, 0u};
  asm volatile("tensor_store_from_lds %0, %1, %2, %3"
               :: "s"(g0), "s"(g1), "s"(gz), "s"(gz) : "memory");
}

// ---- trace of LDS-resident matrix (pitch LD), workgroup-uniform result ------
__device__ __forceinline__ float wg_trace(const float* M, float* red) {
  const int t = threadIdx.x;
  __syncthreads();
  if (t < 128) red[t] = M[t * LD + t];
  __syncthreads();
  if (t < 64) red[t] += red[t + 64];
  __syncthreads();
  if (t < 32) red[t] += red[t + 32];
  __syncthreads();
  if (t < 16) red[t] += red[t + 16];
  __syncthreads();
  if (t < 8) red[t] += red[t + 8];
  __syncthreads();
  if (t < 4) red[t] += red[t + 4];
  __syncthreads();
  if (t < 2) red[t] += red[t + 2];
  __syncthreads();
  if (t == 0) red[0] += red[1];
  __syncthreads();
  float v = red[0];
  __syncthreads();
  return v;
}

// C = scale*(A@B) + diagAdd*I (optional relu). A,B in LDS with pitch LD.
// C written to LDS with pitch cPitch (may alias A or B: full K-reduction
// completes before the barriered writeback).
// Wave w computes output rows [16w,16w+16); 8 tiles of 16x16 in v8f accumulators.
__device__ __forceinline__ void wg_mm(const float* A, const float* B,
                                      float* C, int cPitch,
                                      float scale, float diagAdd, int doRelu) {
  const int tid  = threadIdx.x;
  const int wave = tid >> 5;
  const int lane = tid & 31;
  const int half = lane >> 4;   // lanes 16..31 hold K+2 / M+8 per ISA layout
  const int r    = lane & 15;
  const int m0   = wave << 4;

  v8f acc[8] = {};
  for (int k0 = 0; k0 < 128; k0 += 4) {
    // A fragment 16x4: v0 = (M=r, K=k0+2*half), v1 = K+1   (ISA 32-bit A layout)
    v2f a;
    const float* ap = &A[(m0 + r) * LD + k0 + 2 * half];
    a.x = ap[0];
    a.y = ap[1];
#pragma unroll
    for (int n = 0; n < 8; ++n) {
      // B fragment 4x16: v0 = (K=k0+2*half, N=n0+r), v1 = K+1 row
      v2f bf;
      bf.x = B[(k0 + 2 * half) * LD + (n << 4) + r];
      bf.y = B[(k0 + 2 * half + 1) * LD + (n << 4) + r];
      acc[n] = __builtin_amdgcn_wmma_f32_16x16x4_f32(
          false, a, false, bf, (short)0, acc[n], false, false);
    }
  }
  __syncthreads();  // all waves done reading A/B before we may overwrite them
#pragma unroll
  for (int n = 0; n < 8; ++n) {
#pragma unroll
    for (int j = 0; j < 8; ++j) {
      const int M = m0 + j + 8 * half;   // C/D layout: VGPR j -> M = j (+8 hi lanes)
      const int N = (n << 4) + r;
      float v = acc[n][j] * scale;
      if (M == N) v += diagAdd;
      if (doRelu) v = fmaxf(v, 0.0f);
      C[M * cPitch + N] = v;
    }
  }
  __syncthreads();
}

__device__ __forceinline__ void wg_set_scaled_id(float* Bf, float d) {
  for (int i = threadIdx.x; i < 16384; i += 256) {
    const int row = i >> 7, col = i & 127;
    Bf[row * LD + col] = (row == col) ? d : 0.0f;
  }
  __syncthreads();
}

__global__ __launch_bounds__(256, 1) void relut_spd_kernel(
    const float* __restrict__ Pg, float* __restrict__ Og) {
  __shared__ float bufY[128 * LD];   // -> Gs / sqrt(s)
  __shared__ float bufZ[128 * LD];   // -> Gi * sqrt(s)
  __shared__ float bufU[128 * LD];   // work (holds P initially)
  __shared__ float bufV[128 * LD];   // work (dense result tile at the end)
  __shared__ float red[128];

  const int b   = blockIdx.x;
  const int tid = threadIdx.x;
  const float* P = Pg + (size_t)b * 16384;
  float*       O = Og + (size_t)b * 16384;

  // ---- TDM: DMA P -> bufU with hardware row padding (pitch 132 floats) ----
  if (tid == 0) {
    tdm_load_tile(P, bufU);
    __builtin_amdgcn_s_wait_tensorcnt(0);
  }
  __syncthreads();

  const float trP   = wg_trace(bufU, red);
  const float s     = trP * (1.0f + ALPHA * 128.0f);  // tr(P') >= lambda_max(P'), SPD
  const float inv_s = 1.0f / s;
  const float dsh   = ALPHA * trP * inv_s;            // shifted diagonal term of P'/s

  // Y = P'/s , Z = I  (keep P in bufU for the Gi*P*Gi step)
  for (int i = tid; i < 16384; i += 256) {
    const int row = i >> 7, col = i & 127;
    const float pv = bufU[row * LD + col];
    bufY[row * LD + col] = pv * inv_s + ((row == col) ? dsh : 0.0f);
    bufZ[row * LD + col] = (row == col) ? 1.0f : 0.0f;
  }
  __syncthreads();

  // ---- coupled Newton-Schulz: Y -> (P'/s)^{1/2}, Z -> (P'/s)^{-1/2} ----
#pragma clang loop unroll(disable)
  for (int it = 0; it < NS_ITERS; ++it) {
    wg_mm(bufZ, bufY, bufV, LD, -0.5f, 1.5f, 0);  // T = 1.5I - 0.5 Z@Y
    wg_mm(bufY, bufV, bufY, LD, 1.0f, 0.0f, 0);   // Y = Y@T
    wg_mm(bufV, bufZ, bufZ, LD, 1.0f, 0.0f, 0);   // Z = T@Z
  }
  // Gs = sqrt(s)*Y, Gi = Z/sqrt(s); scale factors folded into the sandwiches below.

  // ---- M = Gi@P@Gi = (Z@P@Z)/s ----
  wg_mm(bufZ, bufU, bufV, LD, 1.0f, 0.0f, 0);     // W = Z@P
  wg_mm(bufV, bufZ, bufU, LD, inv_s, 0.0f, 0);    // M -> bufU (P dead)

  const float trM = wg_trace(bufU, red);
  if (tid < 128) bufU[tid * LD + tid] += ALPHA * trM - 1.0f;  // E = M' - I
  __syncthreads();

  // ---- L = log(I+E) = E@(a1 I + E@(a2 I + ... a8 I)),  a_j = (-1)^{j+1}/j ----
  wg_set_scaled_id(bufV, -1.0f / (float)LOG_TERMS);    // S = a8*I
#pragma clang loop unroll(disable)
  for (int j = LOG_TERMS - 1; j >= 1; --j) {
    const float aj = ((j & 1) ? 1.0f : -1.0f) / (float)j;
    wg_mm(bufU, bufV, bufV, LD, 1.0f, aj, 0);     // S = aj*I + E@S
  }
  wg_mm(bufU, bufV, bufV, LD, 1.0f, 0.0f, 0);     // L = E@S -> bufV

  // ---- r = relu(Gs@L@Gs) = relu(s * Y@L@Y) ----
  wg_mm(bufY, bufV, bufU, LD, 1.0f, 0.0f, 0);     // W = Y@L
  wg_mm(bufU, bufY, bufU, LD, s, 0.0f, 1);        // r -> bufU

  // ---- N = Gi@r@Gi = (Z@r@Z)/s ----
  wg_mm(bufZ, bufU, bufV, LD, 1.0f, 0.0f, 0);     // W = Z@r
  wg_mm(bufV, bufZ, bufU, LD, inv_s, 0.0f, 0);    // N -> bufU

  // ---- expN = I + N@(I + N/2@(...)):  S=I; for j=k..1: S = I + (N@S)/j ----
  wg_set_scaled_id(bufV, 1.0f);
#pragma clang loop unroll(disable)
  for (int j = EXP_TERMS; j >= 1; --j) {
    wg_mm(bufU, bufV, bufV, LD, 1.0f / (float)j, 1.0f, 0);
  }

  // ---- out = Gs@expN@Gs = s * Y@S@Y ----
  wg_mm(bufY, bufV, bufU, LD, 1.0f, 0.0f, 0);     // W = Y@S
  wg_mm(bufU, bufY, bufV, 128, s, 0.0f, 0);       // dense pitch-128 result tile

  // ---- TDM: DMA dense result tile bufV -> global output ----
  if (tid == 0) {
    tdm_store_tile(O, bufV);
    __builtin_amdgcn_s_wait_tensorcnt(0);
  }
}

extern "C" void kernel_launch(void* const* d_in, const int* in_sizes, int n_in,
                              void* d_out, int out_size, void* d_ws, size_t ws_size,
                              hipStream_t stream) {
  (void)n_in; (void)out_size; (void)d_ws; (void)ws_size;
  const float* P = (const float*)d_in[0];
  float*       O = (float*)d_out;
  const int nmat = in_sizes[0] / (128 * 128);   // 64*32 = 2048 matrices
  relut_spd_kernel<<<nmat, 256, 0, stream>>>(P, O);
}